// ExchangeableLayer_84739704750361
// MI455X (gfx1250) — compile-verified
//
#include <hip/hip_runtime.h>

typedef float v2f __attribute__((ext_vector_type(2)));
typedef float v8f __attribute__((ext_vector_type(8)));
typedef float f4  __attribute__((ext_vector_type(4)));

// ---------------- small helpers ----------------
__global__ void zero_f32(float* __restrict__ p, int n) {
  int t = blockIdx.x * blockDim.x + threadIdx.x;
  if (t < n) p[t] = 0.f;
}

// column sums of an [N,16] row-major matrix into out[16] (atomic accumulate).
// 16 consecutive threads cover one row -> fully coalesced reads.
__global__ void colsum16(const float* __restrict__ E, int N, float* __restrict__ out) {
  int t = blockIdx.x * blockDim.x + threadIdx.x;
  int u = t & 15;
  int i0 = t >> 4;
  int stride = (gridDim.x * blockDim.x) >> 4;
  float s = 0.f;
  for (int i = i0; i < N; i += stride) s += E[(size_t)i * 16 + u];
  atomicAdd(&out[u], s);
}

// pool_sc[u] = ssum . Wrp_sc[:,u] + csum . Wcp_sc[:,u]; pool_sp likewise with psum.
// launched with 32 threads: lanes 0-15 -> sc, lanes 16-31 -> sp.
__global__ void pool16(const float* __restrict__ ssum, const float* __restrict__ csum,
                       const float* __restrict__ psum,
                       const float* __restrict__ Wrp_sc, const float* __restrict__ Wcp_sc,
                       const float* __restrict__ Wrp_sp, const float* __restrict__ Wcp_sp,
                       float* __restrict__ pool_sc, float* __restrict__ pool_sp) {
  int t = threadIdx.x;
  int u = t & 15;
  const float* other = (t < 16) ? csum : psum;
  const float* Wrp   = (t < 16) ? Wrp_sc : Wrp_sp;
  const float* Wcp   = (t < 16) ? Wcp_sc : Wcp_sp;
  float acc = 0.f;
#pragma unroll
  for (int k = 0; k < 16; ++k)
    acc += ssum[k] * Wrp[k * 16 + u] + other[k] * Wcp[k * 16 + u];
  ((t < 16) ? pool_sc : pool_sp)[u] = acc;
}

// ---------------- tiny GEMM via WMMA: out[N,16] = E[N,16] @ W[16,16] (+ add[16]) ----
// One wave per 16-row tile. K=16 done as 4 chained V_WMMA_F32_16X16X4_F32.
// A fragment (16x4 f32, MxK): lanes 0-15 = M, VGPR0 holds K=k0 (low half) / k0+2
// (high half), VGPR1 holds k0+1 / k0+3.  B mirrored (KxN, N striped on lanes).
// C/D (16x16 f32): lane = N (per half), VGPR v -> M = v + 8*half.
__global__ void gemm16_wmma(const float* __restrict__ E, const float* __restrict__ W,
                            const float* __restrict__ add, float* __restrict__ out,
                            int N, int ntiles) {
  int wave = threadIdx.x >> 5;
  int lane = threadIdx.x & 31;
  int tile = blockIdx.x * (blockDim.x >> 5) + wave;
  if (tile >= ntiles) return;                // wave-uniform: EXEC stays all-ones
  int half = lane >> 4;
  int n    = lane & 15;                      // N index (and M index for A loads)
  int row0 = tile * 16;

  int arow = row0 + n; if (arow > N - 1) arow = N - 1;   // clamp ragged tail
  const float* erow = E + (size_t)arow * 16;

  float cv[8];
#if __has_builtin(__builtin_amdgcn_wmma_f32_16x16x4_f32)
  v8f c = {};
#pragma unroll
  for (int s = 0; s < 4; ++s) {
    int k0 = s * 4 + 2 * half;
    v2f a; a.x = erow[k0];           a.y = erow[k0 + 1];
    v2f b; b.x = W[k0 * 16 + n];     b.y = W[(k0 + 1) * 16 + n];
    c = __builtin_amdgcn_wmma_f32_16x16x4_f32(
            /*neg_a=*/false, a, /*neg_b=*/false, b,
            /*c_mod=*/(short)0, c, /*reuse_a=*/false, /*reuse_b=*/false);
  }
#pragma unroll
  for (int v = 0; v < 8; ++v) cv[v] = c[v];
#else
  // scalar fallback with identical output layout
#pragma unroll
  for (int v = 0; v < 8; ++v) {
    int m = v + 8 * half;
    int grow = row0 + m; if (grow > N - 1) grow = N - 1;
    float acc = 0.f;
#pragma unroll
    for (int k = 0; k < 16; ++k) acc += E[(size_t)grow * 16 + k] * W[k * 16 + n];
    cv[v] = acc;
  }
#endif
  float av = (add != nullptr) ? add[n] : 0.f;
#pragma unroll
  for (int v = 0; v < 8; ++v) {
    int m = v + 8 * half;
    int grow = row0 + m;
    if (grow < N) out[(size_t)grow * 16 + n] = cv[v] + av;
  }
}

// ---------------- dominant kernel: gather + add + relu, streaming NT stores ------
// 4 lanes per nonzero entry, each lane handles one float4 of the 16 outputs.
// Store addresses are contiguous across lanes -> 512B coalesced NT stores/wave.
__global__ void gather_relu16(const int* __restrict__ rows, const int* __restrict__ cols,
                              const float* __restrict__ rowm, const float* __restrict__ colp,
                              float* __restrict__ out, int nnz) {
  int t = blockIdx.x * blockDim.x + threadIdx.x;
  int e = t >> 2;
  if (e >= nnz) return;
  int c4 = (t & 3) * 4;
  int r  = rows[e];
  int cc = cols[e];
  f4 a = *(const f4*)(rowm + (size_t)r  * 16 + c4);   // L2-resident gather
  f4 b = *(const f4*)(colp + (size_t)cc * 16 + c4);   // L2-resident gather
  f4 v = a + b;
  v.x = fmaxf(v.x, 0.f); v.y = fmaxf(v.y, 0.f);
  v.z = fmaxf(v.z, 0.f); v.w = fmaxf(v.w, 0.f);
  __builtin_nontemporal_store(v, (f4*)(out + (size_t)e * 16 + c4));
}

// ---------------- launch ----------------
extern "C" void kernel_launch(void* const* d_in, const int* in_sizes, int n_in,
                              void* d_out, int out_size, void* d_ws, size_t ws_size,
                              hipStream_t stream) {
  const float* S      = (const float*)d_in[0];
  const float* C      = (const float*)d_in[1];
  const float* P      = (const float*)d_in[2];
  const float* Wsc_r  = (const float*)d_in[3];
  const float* Wsc_c  = (const float*)d_in[4];
  const float* Wsc_rp = (const float*)d_in[5];
  const float* Wsc_cp = (const float*)d_in[6];
  const float* Wsp_r  = (const float*)d_in[7];
  const float* Wsp_c  = (const float*)d_in[8];
  const float* Wsp_rp = (const float*)d_in[9];
  const float* Wsp_cp = (const float*)d_in[10];
  const int* sc_rows  = (const int*)d_in[11];
  const int* sc_cols  = (const int*)d_in[12];
  const int* sp_rows  = (const int*)d_in[13];
  const int* sp_cols  = (const int*)d_in[14];

  int Ns = in_sizes[0] / 16;
  int Nc = in_sizes[1] / 16;
  int Np = in_sizes[2] / 16;
  int nnz_sc = in_sizes[11];
  int nnz_sp = in_sizes[13];

  // workspace layout (floats): ~13 MB total
  float* ws      = (float*)d_ws;
  float* rowm_sc = ws;
  float* rowm_sp = rowm_sc + (size_t)Ns * 16;
  float* colp_sc = rowm_sp + (size_t)Ns * 16;
  float* colp_sp = colp_sc + (size_t)Nc * 16;
  float* ssum    = colp_sp + (size_t)Np * 16;
  float* csum    = ssum + 16;
  float* psum    = csum + 16;
  float* pool_sc = psum + 16;
  float* pool_sp = pool_sc + 16;

  float* out_sc = (float*)d_out;
  float* out_sp = out_sc + (size_t)nnz_sc * 16;

  // 1) zero the 80-float accumulator region (ws is poisoned, not zeroed)
  zero_f32<<<1, 128, 0, stream>>>(ssum, 80);

  // 2) column sums of embedding matrices
  colsum16<<<128, 256, 0, stream>>>(S, Ns, ssum);
  colsum16<<<8,   256, 0, stream>>>(C, Nc, csum);
  colsum16<<<4,   256, 0, stream>>>(P, Np, psum);

  // 3) pooled 16-vectors (folded later into col marginals)
  pool16<<<1, 32, 0, stream>>>(ssum, csum, psum, Wsc_rp, Wsc_cp, Wsp_rp, Wsp_cp,
                               pool_sc, pool_sp);

  // 4) marginal tables via WMMA (one wave per 16-row tile, 8 waves/block)
  int tiles_s = (Ns + 15) / 16;
  int tiles_c = (Nc + 15) / 16;
  int tiles_p = (Np + 15) / 16;
  const int wpb = 256 / 32;
  gemm16_wmma<<<(tiles_s + wpb - 1) / wpb, 256, 0, stream>>>(S, Wsc_r, nullptr, rowm_sc, Ns, tiles_s);
  gemm16_wmma<<<(tiles_s + wpb - 1) / wpb, 256, 0, stream>>>(S, Wsp_r, nullptr, rowm_sp, Ns, tiles_s);
  gemm16_wmma<<<(tiles_c + wpb - 1) / wpb, 256, 0, stream>>>(C, Wsc_c, pool_sc, colp_sc, Nc, tiles_c);
  gemm16_wmma<<<(tiles_p + wpb - 1) / wpb, 256, 0, stream>>>(P, Wsp_c, pool_sp, colp_sp, Np, tiles_p);

  // 5) the memory-bound gather + relu (bulk of the 430 MB traffic)
  long long tsc = (long long)nnz_sc * 4;
  long long tsp = (long long)nnz_sp * 4;
  gather_relu16<<<(unsigned)((tsc + 255) / 256), 256, 0, stream>>>(sc_rows, sc_cols, rowm_sc, colp_sc, out_sc, nnz_sc);
  gather_relu16<<<(unsigned)((tsp + 255) / 256), 256, 0, stream>>>(sp_rows, sp_cols, rowm_sp, colp_sp, out_sp, nnz_sp);
}